// MatrixSquareRoot_62749472194673
// MI455X (gfx1250) — compile-verified
//
#include <hip/hip_runtime.h>
#include <math.h>

typedef __attribute__((ext_vector_type(2))) float v2f;
typedef __attribute__((ext_vector_type(8))) float v8f;

#define DIM     512
#define BATCH   64
#define TILE    64     // output tile per workgroup
#define KCHUNK  32     // K staging depth in LDS
#define A_LDS   36     // LDS row stride for A tile (floats): conflict-free, 16B-aligned rows
#define SBP     160    // LDS row stride for k-pair-interleaved B tile (floats); 160 % 64 == 32

// ---------------------------------------------------------------------------
// Per-batch Frobenius norm: norm[b] = sqrt(sum(x[b]^2))
// ---------------------------------------------------------------------------
__global__ __launch_bounds__(256)
void ns_norm_kernel(const float* __restrict__ x, float* __restrict__ norm)
{
    __shared__ float red[256];
    const int b = blockIdx.x;
    const float* xb = x + (size_t)b * DIM * DIM;
    float s = 0.0f;
    for (int i = threadIdx.x; i < DIM * DIM; i += 256) {
        float v = xb[i];
        s += v * v;
    }
    red[threadIdx.x] = s;
    __syncthreads();
    for (int off = 128; off > 0; off >>= 1) {
        if (threadIdx.x < off) red[threadIdx.x] += red[threadIdx.x + off];
        __syncthreads();
    }
    if (threadIdx.x == 0) norm[b] = sqrtf(red[0]);
}

// ---------------------------------------------------------------------------
// Y = x / norm[b]; Z = I (broadcast per batch)
// ---------------------------------------------------------------------------
__global__ __launch_bounds__(256)
void ns_init_kernel(const float* __restrict__ x, const float* __restrict__ norm,
                    float* __restrict__ Y, float* __restrict__ Z)
{
    size_t i = (size_t)blockIdx.x * 256 + threadIdx.x;
    int b      = (int)(i >> 18);             // / (512*512)
    int within = (int)(i & (DIM * DIM - 1));
    int row = within >> 9;
    int col = within & (DIM - 1);
    float rn = 1.0f / norm[b];
    Y[i] = x[i] * rn;
    Z[i] = (row == col) ? 1.0f : 0.0f;
}

// ---------------------------------------------------------------------------
// Batched GEMM: C[b] = alpha * A[b] @ B[b] + diag * I
// grid = (64 tiles, BATCH), block = 128 threads (4 waves of 32).
// Each wave computes a 32x32 sub-tile via 2x2 V_WMMA_F32_16X16X4_F32 accums.
// B is staged k-pair interleaved so every fragment operand is one b64 load.
// ---------------------------------------------------------------------------
__global__ __launch_bounds__(128)
void ns_gemm_kernel(float* __restrict__ C, const float* __restrict__ A,
                    const float* __restrict__ B, float alpha, float diag)
{
    __shared__ __attribute__((aligned(16))) float sA[TILE * A_LDS];          // 64 x 32, stride 36
    __shared__ __attribute__((aligned(16))) float sB[(KCHUNK / 2) * SBP];    // 16 x (64 pairs), stride 160

    const int b     = blockIdx.y;
    const int tileM = (blockIdx.x >> 3) * TILE;
    const int tileN = (blockIdx.x & 7) * TILE;

    const float* Ab = A + (size_t)b * DIM * DIM;
    const float* Bb = B + (size_t)b * DIM * DIM;
    float*       Cb = C + (size_t)b * DIM * DIM;

    const int t     = threadIdx.x;       // 0..127
    const int lane  = t & 31;
    const int wave  = t >> 5;            // 0..3
    const int waveM = (wave >> 1) * 32;  // 0 or 32
    const int waveN = (wave & 1) * 32;   // 0 or 32
    const int half  = lane >> 4;         // K sub-offset selector (0 -> K{0,1}, 1 -> K{2,3})
    const int lm    = lane & 15;

    v8f acc00 = {}, acc01 = {}, acc10 = {}, acc11 = {};

    for (int kk = 0; kk < DIM; kk += KCHUNK) {
        // ---- stage A tile 64x32: 512 float4, 4 per thread ----
        #pragma unroll
        for (int i = 0; i < 4; ++i) {
            int idx = t + i * 128;           // 0..511
            int row = idx >> 3;              // 0..63
            int c4  = (idx & 7) << 2;        // 0..28
            float4 v = *(const float4*)(Ab + (size_t)(tileM + row) * DIM + kk + c4);
            *(float4*)&sA[row * A_LDS + c4] = v;   // byte offset = row*144 + c4*4 -> 16B aligned
        }
        // ---- stage B tile 32x64, k-pair interleaved: (k,n) -> sB[(k>>1)*SBP + 2n + (k&1)] ----
        #pragma unroll
        for (int i = 0; i < 2; ++i) {
            int idx = t + i * 128;           // 0..255
            int kp  = idx >> 4;              // 0..15 : row pair (2*kp, 2*kp+1)
            int c4  = (idx & 15) << 2;       // 0..60
            const float* src = Bb + (size_t)(kk + 2 * kp) * DIM + tileN + c4;
            float4 r0 = *(const float4*)(src);
            float4 r1 = *(const float4*)(src + DIM);
            float4 o0 = make_float4(r0.x, r1.x, r0.y, r1.y);
            float4 o1 = make_float4(r0.z, r1.z, r0.w, r1.w);
            float* dst = &sB[kp * SBP + 2 * c4];   // byte offset = kp*640 + c4*8 -> 16B aligned
            *(float4*)(dst)     = o0;
            *(float4*)(dst + 4) = o1;
        }
        __syncthreads();

        // Prefetch next K-chunk (emits global_prefetch_b8); uniform branch.
        if (kk + KCHUNK < DIM) {
            __builtin_prefetch(Ab + (size_t)(tileM + (t >> 1)) * DIM + kk + KCHUNK, 0, 3);
            __builtin_prefetch(Bb + (size_t)(kk + KCHUNK + (t >> 2)) * DIM + tileN, 0, 3);
        }

        #pragma unroll
        for (int k = 0; k < KCHUNK; k += 4) {
            const int ka = k + half * 2;     // even; per-half-wave K offset within the quad
            // A fragments: lane = M, contiguous K pair -> b64 loads
            v2f a0 = *(const v2f*)&sA[(waveM +      lm) * A_LDS + ka];
            v2f a1 = *(const v2f*)&sA[(waveM + 16 + lm) * A_LDS + ka];
            // B fragments: interleaved pair (K=ka, ka+1) at column n -> b64 loads
            const float* bbase = &sB[(ka >> 1) * SBP + 2 * (waveN + lm)];
            v2f b0 = *(const v2f*)(bbase);
            v2f b1 = *(const v2f*)(bbase + 32);  // +16 columns -> +32 floats

            acc00 = __builtin_amdgcn_wmma_f32_16x16x4_f32(false, a0, false, b0, (short)0, acc00, false, false);
            acc01 = __builtin_amdgcn_wmma_f32_16x16x4_f32(false, a0, false, b1, (short)0, acc01, false, false);
            acc10 = __builtin_amdgcn_wmma_f32_16x16x4_f32(false, a1, false, b0, (short)0, acc10, false, false);
            acc11 = __builtin_amdgcn_wmma_f32_16x16x4_f32(false, a1, false, b1, (short)0, acc11, false, false);
        }
        __syncthreads();
    }

    // ---- epilogue: C = alpha*acc + diag*I (VGPR r -> rows r and r+8) ----
    #pragma unroll
    for (int r = 0; r < 8; ++r) {
        int row  = tileM + waveM + r + half * 8;
        int col  = tileN + waveN + lm;
        int row1 = row + 16;
        int col1 = col + 16;
        Cb[(size_t)row  * DIM + col ] = alpha * acc00[r] + ((row  == col ) ? diag : 0.0f);
        Cb[(size_t)row  * DIM + col1] = alpha * acc01[r] + ((row  == col1) ? diag : 0.0f);
        Cb[(size_t)row1 * DIM + col ] = alpha * acc10[r] + ((row1 == col ) ? diag : 0.0f);
        Cb[(size_t)row1 * DIM + col1] = alpha * acc11[r] + ((row1 == col1) ? diag : 0.0f);
    }
}

// ---------------------------------------------------------------------------
// out *= sqrt(norm[b])  (in place; out fully rewritten earlier each call)
// ---------------------------------------------------------------------------
__global__ __launch_bounds__(256)
void ns_final_kernel(float* __restrict__ out, const float* __restrict__ norm)
{
    size_t i = (size_t)blockIdx.x * 256 + threadIdx.x;
    int b = (int)(i >> 18);
    out[i] = out[i] * sqrtf(norm[b]);
}

// ---------------------------------------------------------------------------
extern "C" void kernel_launch(void* const* d_in, const int* in_sizes, int n_in,
                              void* d_out, int out_size, void* d_ws, size_t ws_size,
                              hipStream_t stream)
{
    (void)in_sizes; (void)n_in; (void)out_size; (void)ws_size;
    const float* x = (const float*)d_in[0];      // [64,512,512] fp32 (d_in[1] = I, unused)
    float* out = (float*)d_out;                  // [64,512,512] fp32

    const size_t N = (size_t)BATCH * DIM * DIM;  // 16,777,216 elems = 64 MB

    // Workspace layout: norm[64] | Ya | Za | Zb | T  (~256 MB + 256 B)
    float* norm = (float*)d_ws;
    float* Ya = norm + 64;
    float* Za = Ya + N;
    float* Zb = Za + N;
    float* T  = Zb + N;

    ns_norm_kernel<<<BATCH, 256, 0, stream>>>(x, norm);
    ns_init_kernel<<<(unsigned)(N / 256), 256, 0, stream>>>(x, norm, Ya, Za);

    dim3 ggrid(64, BATCH);   // 64 output tiles of 64x64 per 512x512 matrix
    float* Ycur = Ya;  float* Ynext = out;   // d_out is the odd-parity Y buffer
    float* Zcur = Za;  float* Znext = Zb;

    for (int it = 0; it < 5; ++it) {
        // T = 1.5*I - 0.5 * (Z @ Y)
        ns_gemm_kernel<<<ggrid, 128, 0, stream>>>(T, Zcur, Ycur, -0.5f, 1.5f);
        // Y' = Y @ T
        ns_gemm_kernel<<<ggrid, 128, 0, stream>>>(Ynext, Ycur, T, 1.0f, 0.0f);
        // Z' = T @ Z
        ns_gemm_kernel<<<ggrid, 128, 0, stream>>>(Znext, T, Zcur, 1.0f, 0.0f);
        float* tmp = Ycur; Ycur = Ynext; Ynext = tmp;
        tmp = Zcur; Zcur = Znext; Znext = tmp;
    }
    // After 5 (odd) swaps, Y lives in d_out. Apply final sqrt(norm) scale.
    ns_final_kernel<<<(unsigned)(N / 256), 256, 0, stream>>>(out, norm);
}